// MultiHeadModel_75050258530467
// MI455X (gfx1250) — compile-verified
//
#include <hip/hip_runtime.h>
#include <hip/hip_bf16.h>

// Problem constants (fixed by the reference setup).
#define NN 100000
#define GG 1024
#define CC 16
#define DD 256       // D_IN == D_OUT
#define ROWS (GG * CC)   // 16384

typedef float v2f __attribute__((ext_vector_type(2)));
typedef float v8f __attribute__((ext_vector_type(8)));

// ---------------------------------------------------------------------------
// 1) Segment boundaries from the sorted batch array: starts[g] = first node
//    with batch >= g, starts[G] = N.
// ---------------------------------------------------------------------------
__global__ void boundaries_kernel(const int* __restrict__ batch,
                                  int* __restrict__ starts) {
    int n = blockIdx.x * blockDim.x + threadIdx.x;
    if (n >= NN) return;
    if (n == 0) {
        int b = batch[0];
        for (int g = 0; g <= b; ++g) starts[g] = 0;
    } else {
        int b0 = batch[n - 1], b1 = batch[n];
        for (int g = b0 + 1; g <= b1; ++g) starts[g] = n;
    }
    if (n == NN - 1) {
        int b = batch[NN - 1];
        for (int g = b + 1; g <= GG; ++g) starts[g] = NN;
    }
}

// ---------------------------------------------------------------------------
// 2) Masked segment mean of raw features: Mout[g,:] = mean_{n in g, mask} x[n,:]
//    One block (256 threads) per graph; thread t owns feature column t.
//    Skips rows with mask==0 (halves HBM traffic on the dominant pass).
// ---------------------------------------------------------------------------
__global__ void seg_mean_kernel(const float* __restrict__ x,
                                const unsigned char* __restrict__ mask,
                                const int* __restrict__ starts,
                                float* __restrict__ Mout) {
    int g = blockIdx.x;
    int t = threadIdx.x;                // 0..255 (== DD)
    int s = starts[g], e = starts[g + 1];
    float acc = 0.f, cnt = 0.f;
    for (int n = s; n < e; ++n) {
        if (mask[n]) {                  // uniform branch across the block
            acc += x[(size_t)n * DD + t];
            cnt += 1.f;
        }
    }
    Mout[(size_t)g * DD + t] = acc / cnt;
}

// ---------------------------------------------------------------------------
// fp32 WMMA GEMM with LDS-staged B panel (row-pair interleaved layout).
//   C[64 rows x 64 cols per block] = A @ W (+bias, optional relu)
//   blockDim = (32,4): wave ty owns rows [tileM, tileM+16); all 4 waves share
//   the B panel W[:, tileN:tileN+64), staged chunk-by-chunk (KC=32 rows, 8KB).
//
//   LDS layout: row-pair p = k/2 interleaved -> Bs[p*128 + c*2 + (k&1)], so a
//   WMMA B fragment {W[k+2lh][c], W[k+1+2lh][c]} is ONE contiguous 8-byte
//   ds_load_b64 into an even-aligned VGPR pair (no v_mov reassembly).
//
//   V_WMMA_F32_16X16X4_F32 operand layout (ISA 7.12.2):
//     A 16x4 : lane L holds row M=L&15; VGPR0 = K (L<16 ? 0 : 2), VGPR1 = +1
//     B 4x16 : lane L holds col N=L&15; VGPR0 = K (L<16 ? 0 : 2), VGPR1 = +1
//     C/D    : VGPR v, lane L -> (M = v + (L>=16)*8, N = L&15)
// ---------------------------------------------------------------------------
template <int N_, int K_, int LDC>
__global__ __launch_bounds__(128)
void gemm_wmma_lds(const float* __restrict__ A, const float* __restrict__ W,
                   const float* __restrict__ bias, float* __restrict__ Cdst,
                   const int* __restrict__ rowmap, int relu) {
    constexpr int KC = 32;                            // K-chunk per barrier
    __shared__ __align__(16) float Bs[(KC / 2) * 128]; // 8 KB, pair-interleaved

    const int lane = threadIdx.x;        // 0..31 (wave32)
    const int ty   = threadIdx.y;        // 0..3
    const int tid  = ty * 32 + lane;     // 0..127
    const int l15  = lane & 15;
    const int lh   = lane >> 4;          // 0 or 1
    const int tileM = (blockIdx.y * 4 + ty) * 16;
    const int tileN = blockIdx.x * 64;

    int arow = tileM + l15;
    if (rowmap) arow = rowmap[arow];
    const float* Ap = A + (size_t)arow * K_ + 2 * lh;   // 8B-aligned (K_%4==0)

    v8f acc[4] = {};

    for (int k0 = 0; k0 < K_; k0 += KC) {
        // ---- cooperative stage: W[k0:k0+KC, tileN:tileN+64] -> Bs ----
        // Each thread: 8 interleaved float4 stores, fed by 2 coalesced float2
        // loads from consecutive W rows.
        const float* Wc = W + (size_t)k0 * N_ + tileN;
#pragma unroll
        for (int r = 0; r < 8; ++r) {
            int f4 = r * 128 + tid;            // float4 index 0..1023
            int p  = f4 >> 5;                  // row-pair 0..31
            int c  = (f4 & 31) * 2;            // column 0..62 (step 2)
            v2f g0 = *(const v2f*)&Wc[(size_t)(2 * p)     * N_ + c];
            v2f g1 = *(const v2f*)&Wc[(size_t)(2 * p + 1) * N_ + c];
            float4 st;
            st.x = g0.x; st.y = g1.x;          // col c   : rows (2p, 2p+1)
            st.z = g0.y; st.w = g1.y;          // col c+1 : rows (2p, 2p+1)
            *(float4*)&Bs[p * 128 + c * 2] = st;
        }
        __syncthreads();

        // ---- preload A chunk into registers (one loadcnt wait per chunk) ----
        v2f afrag[KC / 4];
#pragma unroll
        for (int i = 0; i < KC / 4; ++i)
            afrag[i] = *(const v2f*)(Ap + k0 + i * 4);

        // ---- 32 WMMAs per chunk; each B fragment = one ds_load_b64 ----
#pragma unroll
        for (int i = 0; i < KC / 4; ++i) {
            // k-step kk = 4*i; B pair index = kk/2 + lh = 2*i + lh
            const float* Bp = &Bs[(2 * i + lh) * 128 + l15 * 2];
#pragma unroll
            for (int t = 0; t < 4; ++t) {
                v2f b = *(const v2f*)(Bp + t * 32);   // {row kk+2lh, kk+1+2lh}
                acc[t] = __builtin_amdgcn_wmma_f32_16x16x4_f32(
                    false, afrag[i], false, b, (short)0, acc[t], false, false);
            }
        }
        __syncthreads();
    }

#pragma unroll
    for (int v = 0; v < 8; ++v) {
        int row = tileM + v + lh * 8;
        float* crow = Cdst + (size_t)row * LDC;
#pragma unroll
        for (int t = 0; t < 4; ++t) {
            int c = tileN + t * 16 + l15;
            float val = acc[t][v] + bias[c];
            if (relu) val = fmaxf(val, 0.f);
            crow[c] = val;
        }
    }
}

// ---------------------------------------------------------------------------
// 5) Expand CE (repeat each graph row C=16 times) into left half of Hin.
// ---------------------------------------------------------------------------
__global__ void expand_ce_kernel(const float* __restrict__ CE,
                                 float* __restrict__ Hin) {
    int idx = blockIdx.x * blockDim.x + threadIdx.x;   // < ROWS*DD
    int r = idx >> 8;
    int c = idx & 255;
    Hin[(size_t)r * (2 * DD) + c] = CE[(size_t)(r >> 4) * DD + c];
}

// ---------------------------------------------------------------------------
// 8) Final head: out[r] = H2[r,:] . W3 + b3  (one wave per row, xor-shuffle)
// ---------------------------------------------------------------------------
__global__ void head_kernel(const float* __restrict__ H2,
                            const float* __restrict__ W3,
                            const float* __restrict__ b3,
                            float* __restrict__ out) {
    int wave = threadIdx.x >> 5;
    int lane = threadIdx.x & 31;
    int row = blockIdx.x * 8 + wave;
    const float* hr = H2 + (size_t)row * DD;
    float p = 0.f;
#pragma unroll
    for (int j = lane; j < DD; j += 32) p += hr[j] * W3[j];
#pragma unroll
    for (int off = 16; off >= 1; off >>= 1) p += __shfl_xor(p, off, 32);
    if (lane == 0) out[row] = p + b3[0];
}

// ---------------------------------------------------------------------------
extern "C" void kernel_launch(void* const* d_in, const int* in_sizes, int n_in,
                              void* d_out, int out_size, void* d_ws, size_t ws_size,
                              hipStream_t stream) {
    const float*         x    = (const float*)d_in[0];
    // d_in[1] (edge_attr) is dead in the reference output -> skipped entirely.
    const int*           batch = (const int*)d_in[2];
    const unsigned char* mask  = (const unsigned char*)d_in[3];  // jnp bool = 1 byte
    const int*           tidx  = (const int*)d_in[4];
    const float*         Wp = (const float*)d_in[5];
    const float*         bp = (const float*)d_in[6];
    const float*         W1 = (const float*)d_in[7];
    const float*         b1 = (const float*)d_in[8];
    const float*         W2 = (const float*)d_in[9];
    const float*         b2 = (const float*)d_in[10];
    const float*         W3 = (const float*)d_in[11];
    const float*         b3 = (const float*)d_in[12];
    float* out = (float*)d_out;

    // Workspace layout (~86 MB total).
    char*  ws     = (char*)d_ws;
    int*   starts = (int*)ws;                                     // (G+1) ints
    float* Mws    = (float*)(ws + 8192);                          // [G, D]
    float* CE     = Mws + (size_t)GG * DD;                        // [G, D]
    float* Hin    = CE + (size_t)GG * DD;                         // [ROWS, 2D]
    float* H1     = Hin + (size_t)ROWS * 2 * DD;                  // [ROWS, 2D]
    float* H2     = H1 + (size_t)ROWS * 2 * DD;                   // [ROWS, D]

    dim3 gblk(32, 4);   // 4 waves; block covers 64 rows x 64 cols

    // 1) segment boundaries
    boundaries_kernel<<<(NN + 255) / 256, 256, 0, stream>>>(batch, starts);

    // 2) masked segment mean of raw x (the only full HBM pass)
    seg_mean_kernel<<<GG, DD, 0, stream>>>(x, mask, starts, Mws);

    // 3) CE = M @ Wp + bp                [1024,256]x[256,256]
    gemm_wmma_lds<DD, DD, DD><<<dim3(DD / 64, GG / 64), gblk, 0, stream>>>(
        Mws, Wp, bp, CE, nullptr, 0);

    // 4) Hin[:,256:512] = x[tidx] @ Wp + bp   (row-gathered A)
    gemm_wmma_lds<DD, DD, 2 * DD><<<dim3(DD / 64, ROWS / 64), gblk, 0, stream>>>(
        x, Wp, bp, Hin + DD, tidx, 0);

    // 5) Hin[:,0:256] = repeat(CE, 16)
    expand_ce_kernel<<<(ROWS * DD) / 256, 256, 0, stream>>>(CE, Hin);

    // 6) H1 = relu(Hin @ W1 + b1)        [16384,512]x[512,512]
    gemm_wmma_lds<2 * DD, 2 * DD, 2 * DD>
        <<<dim3((2 * DD) / 64, ROWS / 64), gblk, 0, stream>>>(
        Hin, W1, b1, H1, nullptr, 1);

    // 7) H2 = relu(H1 @ W2 + b2)         [16384,512]x[512,256]
    gemm_wmma_lds<DD, 2 * DD, DD><<<dim3(DD / 64, ROWS / 64), gblk, 0, stream>>>(
        H1, W2, b2, H2, nullptr, 1);

    // 8) out = H2 @ W3 + b3              [16384,256]x[256,1]
    head_kernel<<<ROWS / 8, 256, 0, stream>>>(H2, W3, b3, out);
}